// QuantumLikeLayer_22127671509166
// MI455X (gfx1250) — compile-verified
//
#include <hip/hip_runtime.h>

// ---------------------------------------------------------------------------
// QuantumLikeLayer on MI455X (gfx1250):
//   out = (H2048 @ x) * CZdiag, batched over 16384 rows.
// Factored Walsh-Hadamard: H2048 = H16(bits10-7) (x) H16(bits6-3) (x) H8(bits2-0)
// executed as three batched 16x16x16 GEMM stages with v_wmma_f32_16x16x4_f32
// (full f32 precision; problem is HBM-bandwidth-bound at ~268MB -> ~11.5us).
// CZ diagonal sign (-1 iff output bits 10&9 set) is folded into stage-1's
// A-matrix rows (stage 1 produces the final bits 10-7).
// Global<->LDS staging uses gfx1250 async-to-LDS ops (ASYNCcnt-tracked,
// no VGPR round trip); signature probed: (AS1 int32x4*, AS3 ptr, off, cpol).
// ---------------------------------------------------------------------------

typedef float v2f __attribute__((ext_vector_type(2)));
typedef float v8f __attribute__((ext_vector_type(8)));

#define NCOL    2048          // 2^11 state size
#define ROWS    16            // batch rows per block (= WMMA N dim)
#define STRIDE  2052          // LDS row stride in floats (pad 4: distinct banks, 16B aligned)
#define THREADS 256           // 8 wave32

#if __has_builtin(__builtin_amdgcn_global_load_async_to_lds_b128)
#define USE_ASYNC_LOAD 1
#endif
#if __has_builtin(__builtin_amdgcn_global_store_async_from_lds_b128)
#define USE_ASYNC_STORE 1
#endif

#if defined(USE_ASYNC_LOAD) || defined(USE_ASYNC_STORE)
// Probe-confirmed parameter type: global side is AS1 pointer to int32x4
// (clang: "__attribute__((__vector_size__(4 * sizeof(int)))) int __device__ *").
typedef int b128_t __attribute__((vector_size(16)));
typedef __attribute__((address_space(1))) b128_t* g128_t;
typedef __attribute__((address_space(3))) b128_t* l128_t;

__device__ __forceinline__ void wait_asynccnt0() {
#if __has_builtin(__builtin_amdgcn_s_wait_asynccnt)
  __builtin_amdgcn_s_wait_asynccnt(0);
#else
  asm volatile("s_wait_asynccnt 0x0" ::: "memory");
#endif
}
#endif

// One 16x16x16 tile-GEMM: D = A * B, in-place on a length-16 "line" of the
// transform axis, for all 16 batch rows.
//   A: 16x4 per chunk; lane m=lane%16 holds rows, K split across VGPR x lane-half.
//   B: 4x16 per chunk; lane n=lane%16 = batch row, K = 4*kk + 2*(lane/16) + {0,1}.
//   D: 8 VGPRs; M = vgpr + 8*(lane/16), N = lane%16.
__device__ __forceinline__ void wht_tile(float* __restrict__ lds,
                                         const v2f* __restrict__ A,
                                         int base, int kstride, int n, int h) {
  v2f b[4];
#pragma unroll
  for (int kk = 0; kk < 4; ++kk) {
    const int k0 = kk * 4 + h * 2;
    b[kk][0] = lds[n * STRIDE + (k0 + 0) * kstride + base];
    b[kk][1] = lds[n * STRIDE + (k0 + 1) * kstride + base];
  }
  v8f acc = {};
#pragma unroll
  for (int kk = 0; kk < 4; ++kk) {
    // 8 args: (neg_a, A, neg_b, B, c_mod, C, reuse_a, reuse_b)
    acc = __builtin_amdgcn_wmma_f32_16x16x4_f32(false, A[kk], false, b[kk],
                                                (short)0, acc, false, false);
  }
#pragma unroll
  for (int j = 0; j < 8; ++j)
    lds[n * STRIDE + (j + 8 * h) * kstride + base] = acc[j];
}

__global__ __launch_bounds__(THREADS)
void whtcz_kernel(const float* __restrict__ x, float* __restrict__ out) {
  extern __shared__ float lds[];                 // ROWS * STRIDE floats (~128.3 KB)
  const int tid  = threadIdx.x;
  const int lane = tid & 31;
  const int wave = tid >> 5;
  const int row0 = blockIdx.x * ROWS;

  // ---- global -> LDS: 16 rows x 2048 f32 ----
#ifdef USE_ASYNC_LOAD
  {
    const float* gbase = x + (size_t)row0 * NCOL;
    for (int i = tid; i < ROWS * (NCOL / 4); i += THREADS) {
      const int r  = i >> 9;                     // NCOL/4 == 512
      const int cv = i & 511;
      __builtin_amdgcn_global_load_async_to_lds_b128(
          (g128_t)(gbase + r * NCOL + cv * 4),
          (l128_t)&lds[r * STRIDE + cv * 4], 0, 0);
    }
    wait_asynccnt0();                            // this wave's async loads done
  }
#else
  {
    const float4* g = (const float4*)(x + (size_t)row0 * NCOL);
    for (int i = tid; i < ROWS * (NCOL / 4); i += THREADS) {
      const int r  = i >> 9;
      const int cv = i & 511;
      *(float4*)&lds[r * STRIDE + cv * 4] = g[r * (NCOL / 4) + cv];
    }
  }
#endif
  __syncthreads();

  // ---- build A-matrix operands in registers (per-lane, from ISA layout) ----
  const int m = lane & 15;                       // M row of A / batch row for B,D
  const int h = lane >> 4;                       // lane half selects K pair
  v2f a1[4], a2[4], a3[4];
#pragma unroll
  for (int kk = 0; kk < 4; ++kk) {
#pragma unroll
    for (int j = 0; j < 2; ++j) {
      const int k = kk * 4 + h * 2 + j;
      // H16[m,k] * 1/4
      const float s16 = (__builtin_popcount(m & k) & 1) ? -0.25f : 0.25f;
      a2[kk][j] = s16;
      // stage 1: fold CZ sign into output rows (final bits 10-7 = m; -1 iff m&12==12)
      a1[kk][j] = ((m & 12) == 12) ? -s16 : s16;
      // stage 3: G = I2 (x) H8 over low-4 index (b0,c2,c1,c0), * 1/(2*sqrt(2))
      a3[kk][j] = ((m ^ k) & 8) ? 0.0f
                : ((__builtin_popcount(m & k & 7) & 1) ? -0.35355339f : 0.35355339f);
    }
  }

  // ---- stage 1: H16 (+CZ) along bits 10-7 (k-stride 128); 128 lines, 16/wave ----
  for (int t = 0; t < 16; ++t) {
    const int bc = wave * 16 + t;                // line id = low 7 bits of column
    wht_tile(lds, a1, bc, 128, m, h);
  }
  __syncthreads();

  // ---- stage 2: H16 along bits 6-3 (k-stride 8); line id = (a, c) ----
  for (int t = 0; t < 16; ++t) {
    const int id = wave * 16 + t;
    wht_tile(lds, a2, (id >> 3) * 128 + (id & 7), 8, m, h);
  }
  __syncthreads();

  // ---- stage 3: I2(x)H8 along low 4 bits (k-stride 1); line id = (a, b3b2b1) ----
  for (int t = 0; t < 16; ++t) {
    const int id = wave * 16 + t;
    wht_tile(lds, a3, (id >> 3) * 128 + (id & 7) * 16, 1, m, h);
  }
  __syncthreads();

  // ---- LDS -> global ----
#ifdef USE_ASYNC_STORE
  {
    float* gbase = out + (size_t)row0 * NCOL;
    for (int i = tid; i < ROWS * (NCOL / 4); i += THREADS) {
      const int r  = i >> 9;
      const int cv = i & 511;
      __builtin_amdgcn_global_store_async_from_lds_b128(
          (g128_t)(gbase + r * NCOL + cv * 4),
          (l128_t)&lds[r * STRIDE + cv * 4], 0, 0);
    }
    wait_asynccnt0();                            // (S_ENDPGM would also wait-idle)
  }
#else
  {
    float4* g = (float4*)(out + (size_t)row0 * NCOL);
    for (int i = tid; i < ROWS * (NCOL / 4); i += THREADS) {
      const int r  = i >> 9;
      const int cv = i & 511;
      g[r * (NCOL / 4) + cv] = *(const float4*)&lds[r * STRIDE + cv * 4];
    }
  }
#endif
}

extern "C" void kernel_launch(void* const* d_in, const int* in_sizes, int n_in,
                              void* d_out, int out_size, void* d_ws, size_t ws_size,
                              hipStream_t stream) {
  (void)n_in; (void)out_size; (void)d_ws; (void)ws_size;
  const float* x  = (const float*)d_in[0];
  float* out      = (float*)d_out;
  const int batch = in_sizes[0] / NCOL;          // 16384
  const size_t shmem = (size_t)ROWS * STRIDE * sizeof(float);  // 131328 B

  // Allow >64KB dynamic LDS (CDNA5 WGP has 320KB). Deterministic, capture-safe.
  (void)hipFuncSetAttribute((const void*)whtcz_kernel,
                            hipFuncAttributeMaxDynamicSharedMemorySize, (int)shmem);

  whtcz_kernel<<<batch / ROWS, THREADS, shmem, stream>>>(x, out);
}